// SpikePropUnit_9878424780979
// MI455X (gfx1250) — compile-verified
//
#include <hip/hip_runtime.h>

// SNN SRM forward: spike[b,n] = ( dot(input_t[b,n,:], tv) + dot(history[b,n,:], te) ) > thr
// B=16, N=2048, T=1000 -> 32768 rows x 1000-long dual dot product. Pure BW-bound
// (262 MB @ 23.3 TB/s ~ 11us). Dot products run on V_WMMA_F32_16X16X4_F32 (exact f32).

typedef float v2f __attribute__((ext_vector_type(2)));
typedef float v8f __attribute__((ext_vector_type(8)));

#define SNN_B   16
#define SNN_N   2048
#define SNN_T   1000
#define SNN_ROWS (SNN_B * SNN_N)   // 32768 outputs

__global__ __launch_bounds__(256) void snn_spike_wmma_kernel(
    const float* __restrict__ input_t,   // [ROWS][T]
    const float* __restrict__ history,   // [ROWS][T]
    const float* __restrict__ tv,        // [T]
    const float* __restrict__ te,        // [T]
    const float* __restrict__ thr_p,     // [1]
    float* __restrict__ out)             // [ROWS]
{
    const int lane  = threadIdx.x & 31;
    const int wave  = (blockIdx.x * blockDim.x + threadIdx.x) >> 5;
    const int row0  = wave << 4;               // 16 rows per wave
    const int m     = lane & 15;               // A-row / B-column owned by this lane
    const int h     = (lane >> 4) << 1;        // K sub-offset: 0 (lanes 0-15) or 2 (lanes 16-31)
    const float bsel = (m == 0) ? 1.0f : 0.0f; // B-matrix: only column 0 carries the template

    const float* rowT = input_t + (size_t)(row0 + m) * SNN_T + h;
    const float* rowH = history + (size_t)(row0 + m) * SNN_T + h;
    const float* tvp  = tv + h;
    const float* tep  = te + h;

    v8f acc = {0.f, 0.f, 0.f, 0.f, 0.f, 0.f, 0.f, 0.f};

    // 250 K-chunks of 4; two WMMAs per chunk share one accumulator:
    // acc = A_hist x B_te + (A_spk x B_tv + acc)
    #pragma unroll 5
    for (int k = 0; k < SNN_T; k += 4) {
        v2f a_t  = *(const v2f*)(rowT + k);   // A tile element pair (8B/lane, aligned)
        v2f a_h  = *(const v2f*)(rowH + k);
        v2f b_tv = *(const v2f*)(tvp + k);    // broadcast template chunk (L0-resident)
        v2f b_te = *(const v2f*)(tep + k);
        b_tv *= bsel;                          // zero every B column except column 0
        b_te *= bsel;
        acc = __builtin_amdgcn_wmma_f32_16x16x4_f32(
                  false, a_t, false, b_tv, (short)0, acc, false, false);
        acc = __builtin_amdgcn_wmma_f32_16x16x4_f32(
                  false, a_h, false, b_te, (short)0, acc, false, false);
    }

    // D column 0: lane 0 holds rows 0..7 (acc[0..7]), lane 16 holds rows 8..15.
    const float thr = *thr_p;
    if (m == 0) {
        const int base = row0 + ((lane >> 4) << 3);
        float4 r0, r1;
        r0.x = acc[0] > thr ? 1.0f : 0.0f;
        r0.y = acc[1] > thr ? 1.0f : 0.0f;
        r0.z = acc[2] > thr ? 1.0f : 0.0f;
        r0.w = acc[3] > thr ? 1.0f : 0.0f;
        r1.x = acc[4] > thr ? 1.0f : 0.0f;
        r1.y = acc[5] > thr ? 1.0f : 0.0f;
        r1.z = acc[6] > thr ? 1.0f : 0.0f;
        r1.w = acc[7] > thr ? 1.0f : 0.0f;
        *(float4*)(out + base)     = r0;
        *(float4*)(out + base + 4) = r1;
    }
}

extern "C" void kernel_launch(void* const* d_in, const int* in_sizes, int n_in,
                              void* d_out, int out_size, void* d_ws, size_t ws_size,
                              hipStream_t stream) {
    // setup_inputs() order:
    // 0: input_w (unused in forward)  1: input_t  2: history_output
    // 3: template_tv  4: template_dv (unused)  5: template_te  6: template_de (unused)
    // 7: threshold
    const float* input_t = (const float*)d_in[1];
    const float* history = (const float*)d_in[2];
    const float* tv      = (const float*)d_in[3];
    const float* te      = (const float*)d_in[5];
    const float* thr     = (const float*)d_in[7];
    float* out           = (float*)d_out;

    // 32768 rows / 16 rows-per-wave = 2048 waves; 8 waves (256 threads) per block.
    const int threads = 256;
    const int blocks  = SNN_ROWS / 16 / (threads / 32);   // 256
    snn_spike_wmma_kernel<<<blocks, threads, 0, stream>>>(
        input_t, history, tv, te, thr, out);
}